// TGCN_5437428597509
// MI455X (gfx1250) — compile-verified
//
#include <hip/hip_runtime.h>
#include <hip/hip_bf16.h>

// ---------------------------------------------------------------------------
// T-GCN step on MI455X (gfx1250, wave32).
//   GC1 = relu(gcn(X, W1));  GC2 = sigmoid(gcn(GC1, W2));
//   Z = sig([GC2,H]Wz), R = sig([GC2,H]Wr), Ht = tanh([GC2,H*R]Wh)
//   out = Z*H + (1-Z)*Ht
// Dense GEMMs use V_WMMA_F32_16X16X4_F32 (fp32 WMMA, exact math).
// Sparse aggregation is the bandwidth bottleneck (~1.8 GB): 64 threads per
// edge -> fully coalesced 256B gather + 256B atomic scatter per edge.
// ---------------------------------------------------------------------------

typedef float v2f __attribute__((ext_vector_type(2)));
typedef float v8f __attribute__((ext_vector_type(8)));

#define NODES 100000
#define CH 64

__device__ __forceinline__ v8f wmma_f32_k4(v2f a, v2f b, v8f c) {
  // D(16x16,f32) = A(16x4,f32) * B(4x16,f32) + C
  return __builtin_amdgcn_wmma_f32_16x16x4_f32(
      /*neg_a=*/false, a, /*neg_b=*/false, b,
      /*c_mod=*/(short)0, c, /*reuse_a=*/false, /*reuse_b=*/false);
}

__device__ __forceinline__ float sigmoidf_(float x) {
  return 1.0f / (1.0f + __expf(-x));
}

// ---- degree / normalization -----------------------------------------------

__global__ void init_deg_kernel(float* __restrict__ deg, int n) {
  int i = blockIdx.x * blockDim.x + threadIdx.x;
  if (i < n) deg[i] = 1.0f;  // self-loop weight
}

__global__ void accum_deg_kernel(const int* __restrict__ col,
                                 const float* __restrict__ w,
                                 float* __restrict__ deg, int e) {
  int i = blockIdx.x * blockDim.x + threadIdx.x;
  if (i < e) atomicAdd(&deg[col[i]], w[i]);
}

__global__ void rsqrt_deg_kernel(float* __restrict__ deg, int n) {
  int i = blockIdx.x * blockDim.x + threadIdx.x;
  if (i < n) {
    float d = deg[i];
    deg[i] = (d > 0.0f) ? rsqrtf(d) : 0.0f;  // becomes dinv in place
  }
}

__global__ void edge_norm_kernel(const int* __restrict__ row,
                                 const int* __restrict__ col,
                                 const float* __restrict__ w,
                                 const float* __restrict__ dinv,
                                 float* __restrict__ norm, int e) {
  int i = blockIdx.x * blockDim.x + threadIdx.x;
  if (i < e) norm[i] = dinv[row[i]] * w[i] * dinv[col[i]];
}

// ---- dense GEMM: out[N x 64] = A[N x 64] @ W[64 x 64] via fp32 WMMA -------
// Block = 128 threads (4 waves). Block owns a 16-row strip; wave w owns the
// 16-column tile n0 = 16*w. K stepped by 4 -> 16 WMMAs per wave.
// A-frag (16x4 f32): lanes 0-15 hold K={k,k+1}, lanes 16-31 hold K={k+2,k+3}.
// B-frag (4x16 f32): mirrored over K; C/D: VGPR v -> rows v (lo) / v+8 (hi).

__global__ void gemm64_wmma_kernel(const float* __restrict__ A,
                                   const float* __restrict__ W,
                                   float* __restrict__ out) {
  const int wave = threadIdx.x >> 5;
  const int lane = threadIdx.x & 31;
  const int l15 = lane & 15;
  const int khalf = (lane >> 4) << 1;   // 0 or 2
  const int m0 = blockIdx.x * 16;       // N == 6250*16, no tail
  const int n0 = wave * 16;

  const float* Arow = A + (size_t)(m0 + l15) * CH;
  v8f acc = {};
#pragma unroll
  for (int k = 0; k < CH; k += 4) {
    v2f a, b;
    a.x = Arow[k + khalf];
    a.y = Arow[k + khalf + 1];
    b.x = W[(size_t)(k + khalf) * CH + n0 + l15];
    b.y = W[(size_t)(k + khalf + 1) * CH + n0 + l15];
    acc = wmma_f32_k4(a, b, acc);
  }
  const int srow = m0 + ((lane >> 4) << 3);
  const int scol = n0 + l15;
#pragma unroll
  for (int v = 0; v < 8; ++v)
    out[(size_t)(srow + v) * CH + scol] = acc[v];
}

// ---- edge scatter: agg[col] += norm * h[row], 64 threads per edge ---------

__global__ void scatter_kernel(const int* __restrict__ row,
                               const int* __restrict__ col,
                               const float* __restrict__ norm,
                               const float* __restrict__ h,
                               float* __restrict__ agg, int e) {
  long long t = (long long)blockIdx.x * blockDim.x + threadIdx.x;
  long long eid = t >> 6;
  if (eid >= e) return;
  int c = (int)(t & 63);
  int r = row[eid];
  int cl = col[eid];
  float nv = norm[eid];
  atomicAdd(&agg[(size_t)cl * CH + c], nv * h[(size_t)r * CH + c]);
}

// ---- finalize: out = act(agg + dinv^2 * h + bias) -------------------------

__global__ void finalize_kernel(const float* __restrict__ agg,
                                const float* __restrict__ h,
                                const float* __restrict__ dinv,
                                const float* __restrict__ bias,
                                float* __restrict__ out, int n, int act) {
  long long t = (long long)blockIdx.x * blockDim.x + threadIdx.x;
  if (t >= (long long)n * CH) return;
  int i = (int)(t >> 6);
  int c = (int)(t & 63);
  float di = dinv[i];
  float v = agg[t] + di * di * h[t] + bias[c];
  out[t] = (act == 0) ? fmaxf(v, 0.0f) : sigmoidf_(v);
}

// ---- GRU gate: 3 WMMA GEMMs (K=128) + sigmoid/tanh fusion -----------------
// Block = 128 threads (4 waves) owns 16 node rows; wave w owns cols 16w..16w+15.
// Z,R tiles go to LDS so every wave sees the full 64 R columns for the
// second K-half of the H-tilde GEMM.

__global__ void gru_wmma_kernel(const float* __restrict__ GC,
                                const float* __restrict__ Hp,
                                const float* __restrict__ Wz,
                                const float* __restrict__ bz,
                                const float* __restrict__ Wr,
                                const float* __restrict__ br,
                                const float* __restrict__ Wh,
                                const float* __restrict__ bh,
                                float* __restrict__ out) {
  __shared__ float z_s[16 * CH];
  __shared__ float r_s[16 * CH];

  const int wave = threadIdx.x >> 5;
  const int lane = threadIdx.x & 31;
  const int l15 = lane & 15;
  const int khalf = (lane >> 4) << 1;  // 0 or 2
  const int m0 = blockIdx.x * 16;
  const int n0 = wave * 16;

  const float* GCrow = GC + (size_t)(m0 + l15) * CH;
  const float* Hrow = Hp + (size_t)(m0 + l15) * CH;

  // Z and R: [GC2, H] @ Wz / Wr, K = 128
  v8f accz = {};
  v8f accr = {};
#pragma unroll
  for (int k = 0; k < 2 * CH; k += 4) {
    int kk = k + khalf;  // kk,kk+1 stay on one side of the K=64 boundary
    const float* src = (kk < CH) ? GCrow : (Hrow - CH);
    v2f a, b;
    a.x = src[kk];
    a.y = src[kk + 1];
    b.x = Wz[(size_t)kk * CH + n0 + l15];
    b.y = Wz[(size_t)(kk + 1) * CH + n0 + l15];
    accz = wmma_f32_k4(a, b, accz);
    b.x = Wr[(size_t)kk * CH + n0 + l15];
    b.y = Wr[(size_t)(kk + 1) * CH + n0 + l15];
    accr = wmma_f32_k4(a, b, accr);
  }

  const int srow16 = (lane >> 4) << 3;  // 0 or 8
  const int scol = n0 + l15;
  const float bzc = bz[scol];
  const float brc = br[scol];
#pragma unroll
  for (int v = 0; v < 8; ++v) {
    z_s[(srow16 + v) * CH + scol] = sigmoidf_(accz[v] + bzc);
    r_s[(srow16 + v) * CH + scol] = sigmoidf_(accr[v] + brc);
  }
  __syncthreads();

  // H_tilde: [GC2, H * R] @ Wh
  v8f acch = {};
#pragma unroll
  for (int k = 0; k < 2 * CH; k += 4) {
    int kk = k + khalf;
    v2f a, b;
    if (kk < CH) {
      a.x = GCrow[kk];
      a.y = GCrow[kk + 1];
    } else {
      int kc = kk - CH;
      a.x = Hrow[kc] * r_s[l15 * CH + kc];
      a.y = Hrow[kc + 1] * r_s[l15 * CH + kc + 1];
    }
    b.x = Wh[(size_t)kk * CH + n0 + l15];
    b.y = Wh[(size_t)(kk + 1) * CH + n0 + l15];
    acch = wmma_f32_k4(a, b, acch);
  }

  const float bhc = bh[scol];
#pragma unroll
  for (int v = 0; v < 8; ++v) {
    int r16 = srow16 + v;
    float ht = tanhf(acch[v] + bhc);
    float z = z_s[r16 * CH + scol];
    float hp = Hp[(size_t)(m0 + r16) * CH + scol];
    out[(size_t)(m0 + r16) * CH + scol] = z * hp + (1.0f - z) * ht;
  }
}

// ---------------------------------------------------------------------------

extern "C" void kernel_launch(void* const* d_in, const int* in_sizes, int n_in,
                              void* d_out, int out_size, void* d_ws, size_t ws_size,
                              hipStream_t stream) {
  const float* X = (const float*)d_in[0];
  const int* ei = (const int*)d_in[1];  // [2, E] edge_index (integer buffer)
  const float* ew = (const float*)d_in[2];
  const float* H = (const float*)d_in[3];
  const float* W1 = (const float*)d_in[4];
  const float* b1 = (const float*)d_in[5];
  const float* W2 = (const float*)d_in[6];
  const float* b2 = (const float*)d_in[7];
  const float* Wz = (const float*)d_in[8];
  const float* bz = (const float*)d_in[9];
  const float* Wr = (const float*)d_in[10];
  const float* br = (const float*)d_in[11];
  const float* Wh = (const float*)d_in[12];
  const float* bh = (const float*)d_in[13];
  float* out = (float*)d_out;

  const int N = in_sizes[0] / CH;  // 100000
  const int E = in_sizes[2];       // 1600000
  const int* row = ei;
  const int* col = ei + E;

  // Workspace layout (floats): dinv[N] | norm[E] | h[N*64] | agg[N*64] | GC[N*64]
  float* ws = (float*)d_ws;
  float* dinv = ws;
  float* norm = dinv + N;
  float* h = norm + E;
  float* agg = h + (size_t)N * CH;
  float* GC = agg + (size_t)N * CH;

  const int T = 256;
  const int gN = (N + T - 1) / T;
  const int gE = (E + T - 1) / T;
  const int gNC = (int)(((long long)N * CH + T - 1) / T);
  const int gEC = (int)(((long long)E * CH + T - 1) / T);
  const int gM = N / 16;  // 6250 WMMA row-tiles

  // Symmetric normalization (shared by both conv layers)
  init_deg_kernel<<<gN, T, 0, stream>>>(dinv, N);
  accum_deg_kernel<<<gE, T, 0, stream>>>(col, ew, dinv, E);
  rsqrt_deg_kernel<<<gN, T, 0, stream>>>(dinv, N);
  edge_norm_kernel<<<gE, T, 0, stream>>>(row, col, ew, dinv, norm, E);

  // Layer 1: h = X@W1; agg = scatter; GC = relu(agg + dinv^2*h + b1)
  gemm64_wmma_kernel<<<gM, 128, 0, stream>>>(X, W1, h);
  hipMemsetAsync(agg, 0, (size_t)N * CH * sizeof(float), stream);
  scatter_kernel<<<gEC, T, 0, stream>>>(row, col, norm, h, agg, E);
  finalize_kernel<<<gNC, T, 0, stream>>>(agg, h, dinv, b1, GC, N, /*relu*/ 0);

  // Layer 2: h = GC@W2; agg = scatter; GC = sigmoid(agg + dinv^2*h + b2)
  gemm64_wmma_kernel<<<gM, 128, 0, stream>>>(GC, W2, h);
  hipMemsetAsync(agg, 0, (size_t)N * CH * sizeof(float), stream);
  scatter_kernel<<<gEC, T, 0, stream>>>(row, col, norm, h, agg, E);
  finalize_kernel<<<gNC, T, 0, stream>>>(agg, h, dinv, b2, GC, N, /*sigmoid*/ 1);

  // GRU gate
  gru_wmma_kernel<<<gM, 128, 0, stream>>>(GC, H, Wz, bz, Wr, br, Wh, bh, out);
}